// FeatureNet_29996051595306
// MI455X (gfx1250) — compile-verified
//
#include <hip/hip_runtime.h>
#include <math.h>

// ---------------------------------------------------------------------------
// IPA forward for MI455X (gfx1250, wave32).
//  * All matmuls: V_WMMA_F32_16X16X4_F32 (fp32; workload is HBM-bound on the
//    537MB pair tensor -> full precision costs nothing).
//  * The two 512MB z-streaming GEMMs stage operands into LDS with
//    GLOBAL_LOAD_ASYNC_TO_LDS_B128 (ASYNCcnt) -- double-buffered in k_opair,
//    wave-local in k_logits.
//
// Input order (setup_inputs() dict, recursive insertion order):
//  0 residue_idx (i32 1024)  1 coordinates (f32 1024*4*3)  2 residue_mask
//  3 relpos_w 4 relpos_b  5 reldist_w 6 reldist_b  7/8 pair_ln  9/10 single_ln
// 11/12 q  13/14 kv  15/16 q_pts  17/18 kv_pts  19/20 b  21 head_w  22/23 out
//
// Output: s (1024*384) then pair (1024*1024*128), fp32.  Workspace ~72MB.
// ---------------------------------------------------------------------------

#define NRES   1024
#define CZ     128
#define CS     384
#define NH     12
#define CIPA   16
#define PQK    4
#define PV     8
#define PROJW  1152   // 192 + 384 + 144 + 432
#define CATW   2112   // 192 + 96*4 + 1536
#define INFV   100000.0f

typedef float v2f __attribute__((ext_vector_type(2)));
typedef float v8f __attribute__((ext_vector_type(8)));

__device__ inline v2f mk2(float x, float y) { v2f r; r.x = x; r.y = y; return r; }
__device__ inline v8f zero8() { v8f z = {0.f,0.f,0.f,0.f,0.f,0.f,0.f,0.f}; return z; }

// D = A*B + C, fp32 16x16, K=4
__device__ inline v8f wmma4(v2f a, v2f b, v8f c) {
  return __builtin_amdgcn_wmma_f32_16x16x4_f32(false, a, false, b, (short)0, c,
                                               false, false);
}

// --- CDNA5 async global->LDS DMA (ASYNCcnt tracked), inline asm is portable
// across ROCm7.2 / amdgpu-toolchain.  lds_off = addrspace(3) byte offset
// (low 32 bits of a generic pointer into __shared__, per the LDS aperture
// mapping: flat addr[31:0] == LDS offset).
__device__ inline void async_ld_b128(unsigned lds_off, const void* gptr) {
  asm volatile("global_load_async_to_lds_b128 %0, %1, off"
               :: "v"(lds_off), "v"((unsigned long long)(uintptr_t)gptr)
               : "memory");
}
__device__ inline void wait_async0() {
  asm volatile("s_wait_asynccnt 0x0" ::: "memory");
}
__device__ inline void wait_async2() {
  asm volatile("s_wait_asynccnt 0x2" ::: "memory");
}
__device__ inline unsigned lds_off_of(const void* p) {
  return (unsigned)(uintptr_t)p;
}

__device__ inline float wave_sum(float v) {
  for (int o = 16; o > 0; o >>= 1) v += __shfl_xor(v, o, 32);
  return v;
}

// ============================================================== K1: pair / z
__global__ void k_pair(const int* __restrict__ ridx,
                       const float* __restrict__ coords,
                       const float* __restrict__ mask,
                       const float* __restrict__ rp_w, const float* __restrict__ rp_b,
                       const float* __restrict__ rd_w, const float* __restrict__ rd_b,
                       const float* __restrict__ ln_g, const float* __restrict__ ln_b,
                       float* __restrict__ zout) {
  int wid = (blockIdx.x * blockDim.x + threadIdx.x) >> 5;
  int lane = threadIdx.x & 31;
  int i = wid >> 10, j = wid & (NRES - 1);

  int d = ridx[i] - ridx[j];
  int bin = min(max(d, -32), 32) + 32;
  const float* cai = coords + (i * 4 + 1) * 3;
  const float* caj = coords + (j * 4 + 1) * 3;
  float dx = cai[0] - caj[0], dy = cai[1] - caj[1], dz = cai[2] - caj[2];
  float d2 = dx * dx + dy * dy + dz * dz;
  int am = 0; float best = fabsf(d2 - 3.375f);
  for (int k = 1; k < 15; ++k) {
    float vb = 3.375f + (18.0f / 14.0f) * (float)k;
    float t = fabsf(d2 - vb);
    if (t < best) { best = t; am = k; }
  }
  float pm = mask[i] * mask[j];

  float v[4];
  #pragma unroll
  for (int u = 0; u < 4; ++u) {
    int c = lane + 32 * u;
    if (c < 64) v[u] = rp_w[bin * 64 + c] + rp_b[c];
    else { int cc = c - 64; v[u] = (rd_w[am * 64 + cc] + rd_b[cc]) * pm; }
  }
  float s = v[0] + v[1] + v[2] + v[3];
  float mean = wave_sum(s) * (1.0f / 128.0f);
  float q = 0.f;
  #pragma unroll
  for (int u = 0; u < 4; ++u) { float t = v[u] - mean; q += t * t; }
  float var = wave_sum(q) * (1.0f / 128.0f);
  float inv = rsqrtf(var + 1e-5f);
  float* zr = zout + ((size_t)i * NRES + j) * CZ;
  #pragma unroll
  for (int u = 0; u < 4; ++u) {
    int c = lane + 32 * u;
    zr[c] = (v[u] - mean) * inv * ln_g[c] + ln_b[c];
  }
}

// ====================================================== K2: sinusoid + frames
__global__ void k_sin_frames(const int* __restrict__ ridx,
                             const float* __restrict__ coords,
                             float* __restrict__ s_init,
                             float* __restrict__ Rb, float* __restrict__ tb) {
  int n = blockIdx.x, t = threadIdx.x;
  if (t < 192) {
    float inv = expf(-(2.0f * (float)t / (float)CS) * 9.210340371976184f);
    float ang = (float)ridx[n] * inv;
    s_init[(size_t)n * CS + t] = sinf(ang);
    s_init[(size_t)n * CS + 192 + t] = cosf(ang);
  }
  if (t == 0) {
    const float* pn = coords + (n * 4 + 0) * 3;
    const float* ca = coords + (n * 4 + 1) * 3;
    const float* pc = coords + (n * 4 + 2) * 3;
    float e0[3], e1[3], e2[3];
    float s0 = 0.f;
    for (int x = 0; x < 3; ++x) { e0[x] = ca[x] - pn[x]; s0 += e0[x] * e0[x]; }
    float r0 = rsqrtf(s0 + 1e-12f);
    for (int x = 0; x < 3; ++x) e0[x] *= r0;
    float dp = 0.f;
    for (int x = 0; x < 3; ++x) { e1[x] = pc[x] - ca[x]; dp += e1[x] * e0[x]; }
    float s1 = 0.f;
    for (int x = 0; x < 3; ++x) { e1[x] -= dp * e0[x]; s1 += e1[x] * e1[x]; }
    float r1 = rsqrtf(s1 + 1e-12f);
    for (int x = 0; x < 3; ++x) e1[x] *= r1;
    e2[0] = e0[1] * e1[2] - e0[2] * e1[1];
    e2[1] = e0[2] * e1[0] - e0[0] * e1[2];
    e2[2] = e0[0] * e1[1] - e0[1] * e1[0];
    for (int x = 0; x < 3; ++x) {
      Rb[n * 9 + x * 3 + 0] = e0[x];
      Rb[n * 9 + x * 3 + 1] = e1[x];
      Rb[n * 9 + x * 3 + 2] = e2[x];
      tb[n * 3 + x] = ca[x];
    }
  }
}

// ================================================= K3: fused projection GEMM
__global__ void k_proj(const float* __restrict__ s,
                       const float* __restrict__ wq,  const float* __restrict__ bq,
                       const float* __restrict__ wkv, const float* __restrict__ bkv,
                       const float* __restrict__ wqp, const float* __restrict__ bqp,
                       const float* __restrict__ wkp, const float* __restrict__ bkp,
                       float* __restrict__ proj) {
  int i0 = blockIdx.x * 16, c0 = blockIdx.y * 16;
  int lane = threadIdx.x & 31, m = lane & 15, half = lane >> 4, n = m;
  const float* W; const float* bias; int ld, co;
  if (c0 < 192)      { W = wq;  bias = bq;  ld = 192; co = c0; }
  else if (c0 < 576) { W = wkv; bias = bkv; ld = 384; co = c0 - 192; }
  else if (c0 < 720) { W = wqp; bias = bqp; ld = 144; co = c0 - 576; }
  else               { W = wkp; bias = bkp; ld = 432; co = c0 - 720; }
  const float* Arow = s + (size_t)(i0 + m) * CS;
  v8f acc = zero8();
  for (int k = 0; k < CS; k += 4) {
    v2f a = mk2(Arow[k + 2 * half], Arow[k + 2 * half + 1]);
    v2f b = mk2(W[(size_t)(k + 2 * half) * ld + co + n],
                W[(size_t)(k + 2 * half + 1) * ld + co + n]);
    acc = wmma4(a, b, acc);
  }
  #pragma unroll
  for (int r = 0; r < 8; ++r) {
    int row = r + 8 * half;
    proj[(size_t)(i0 + row) * PROJW + c0 + n] = acc[r] + bias[co + n];
  }
}

// ============================================= K4: split q/k/v, rotate points
__global__ void k_split(const float* __restrict__ proj,
                        const float* __restrict__ Rb, const float* __restrict__ tb,
                        float* __restrict__ qh, float* __restrict__ kh,
                        float* __restrict__ vh,
                        float* __restrict__ qg, float* __restrict__ kg,
                        float* __restrict__ vg,
                        float* __restrict__ qn, float* __restrict__ kn) {
  int n = blockIdx.x, t = threadIdx.x;
  const float* pr = proj + (size_t)n * PROJW;
  const float* R = Rb + n * 9;
  const float* tv = tb + n * 3;
  if (t < 192) {
    int h = t >> 4, ch = t & 15;
    qh[((size_t)h * NRES + n) * CIPA + ch] = pr[h * 16 + ch];
    kh[((size_t)h * NRES + n) * CIPA + ch] = pr[192 + h * 32 + ch];
    vh[((size_t)h * NRES + n) * CIPA + ch] = pr[192 + h * 32 + 16 + ch];
  }
  if (t < 144) {  // kv_pts: h = t/12, p = t%12
    int h = t / 12, p = t % 12;
    float l0 = pr[720 + 0 * 144 + h * 12 + p];
    float l1 = pr[720 + 1 * 144 + h * 12 + p];
    float l2 = pr[720 + 2 * 144 + h * 12 + p];
    for (int x = 0; x < 3; ++x) {
      float g = R[x * 3 + 0] * l0 + R[x * 3 + 1] * l1 + R[x * 3 + 2] * l2 + tv[x];
      if (p < PQK) kg[((size_t)h * NRES + n) * 12 + p * 3 + x] = g;
      else         vg[((size_t)h * NRES + n) * 24 + (p - PQK) * 3 + x] = g;
    }
  } else if (t < 192) {  // q_pts
    int u = t - 144, h = u >> 2, p = u & 3;
    float l0 = pr[576 + 0 * 48 + h * 4 + p];
    float l1 = pr[576 + 1 * 48 + h * 4 + p];
    float l2 = pr[576 + 2 * 48 + h * 4 + p];
    for (int x = 0; x < 3; ++x) {
      float g = R[x * 3 + 0] * l0 + R[x * 3 + 1] * l1 + R[x * 3 + 2] * l2 + tv[x];
      qg[((size_t)h * NRES + n) * 12 + p * 3 + x] = g;
    }
  }
  __syncthreads();
  if (t < 24) {
    int h = t >> 1;
    if ((t & 1) == 0) {
      float s = 0.f;
      for (int e = 0; e < 12; ++e) {
        float q = qg[((size_t)h * NRES + n) * 12 + e]; s += q * q;
      }
      qn[(size_t)h * NRES + n] = s;
    } else {
      float s = 0.f;
      for (int e = 0; e < 12; ++e) {
        float q = kg[((size_t)h * NRES + n) * 12 + e]; s += q * q;
      }
      kn[(size_t)h * NRES + n] = s;
    }
  }
}

// ===================================== K5a: logits (bias GEMM + qk + pt + mask)
// z tile for the bias GEMM is staged wave-locally with async DMA into LDS.
__global__ void k_logits(const float* __restrict__ qh, const float* __restrict__ kh,
                         const float* __restrict__ qg, const float* __restrict__ kg,
                         const float* __restrict__ qn, const float* __restrict__ kn,
                         const float* __restrict__ z,
                         const float* __restrict__ wb, const float* __restrict__ bb,
                         const float* __restrict__ head_w,
                         const float* __restrict__ mask,
                         float* __restrict__ abuf) {
  __shared__ float lds_bias[256 * 16];        // [q_local*16 + k_local][head]
  __shared__ float zstage[4][16 * CZ];        // per-wave 8KB z tile
  int i0 = blockIdx.x * 16, j0 = blockIdx.y * 16;
  int wv = threadIdx.x >> 5;
  int lane = threadIdx.x & 31, m = lane & 15, half = lane >> 4, n = m;

  // ---- bias tiles: rows = (q_local=rt, k_local), K = C_Z, N = 12 heads
  float* zs = &zstage[wv][0];
  unsigned zs_base = lds_off_of(zs);
  for (int rt = wv; rt < 16; rt += 4) {
    // async-DMA z[(i0+rt)][j0+0..15][0..127] into this wave's LDS tile:
    // 512 x 16B packets, 16 per lane, contiguous 512B per issue group.
    const float* gsrc = z + ((size_t)(i0 + rt) * NRES + j0) * CZ;
    #pragma unroll
    for (int u = 0; u < 16; ++u) {
      int f = lane + 32 * u;             // f in [0,512): row=f>>5, seg=f&31
      async_ld_b128(zs_base + (unsigned)f * 16, (const char*)gsrc + f * 16);
    }
    wait_async0();   // producer == consumer wave: no barrier needed

    v8f acc = zero8();
    for (int k = 0; k < CZ; k += 4) {
      v2f a = mk2(zs[m * CZ + k + 2 * half], zs[m * CZ + k + 2 * half + 1]);
      v2f b = (n < NH) ? mk2(wb[(k + 2 * half) * NH + n],
                             wb[(k + 2 * half + 1) * NH + n])
                       : mk2(0.f, 0.f);
      acc = wmma4(a, b, acc);
    }
    #pragma unroll
    for (int r = 0; r < 8; ++r) {
      int row = r + 8 * half;  // k_local
      lds_bias[(rt * 16 + row) * 16 + n] = acc[r] + ((n < NH) ? bb[n] : 0.f);
    }
  }
  __syncthreads();

  const float s1 = 0.14433756729740643f;  // sqrt(1/(3*16))
  const float s2 = 0.57735026918962576f;  // sqrt(1/3)
  for (int h = wv; h < NH; h += 4) {
    float hw = log1pf(expf(head_w[h])) * 0.13608276348795434f;  // sqrt(1/54)
    v8f c1 = zero8(), c2 = zero8();
    const float* qrow = qh + ((size_t)h * NRES + i0 + m) * CIPA;
    const float* krow = kh + ((size_t)h * NRES + j0 + n) * CIPA;
    for (int k = 0; k < CIPA; k += 4) {
      v2f a = mk2(qrow[k + 2 * half], qrow[k + 2 * half + 1]);
      v2f b = mk2(krow[k + 2 * half], krow[k + 2 * half + 1]);
      c1 = wmma4(a, b, c1);
    }
    const float* qp = qg + ((size_t)h * NRES + i0 + m) * 12;
    const float* kp = kg + ((size_t)h * NRES + j0 + n) * 12;
    for (int k = 0; k < 12; k += 4) {
      v2f a = mk2(qp[k + 2 * half], qp[k + 2 * half + 1]);
      v2f b = mk2(kp[k + 2 * half], kp[k + 2 * half + 1]);
      c2 = wmma4(a, b, c2);
    }
    #pragma unroll
    for (int r = 0; r < 8; ++r) {
      int row = r + 8 * half;
      int gi = i0 + row, gj = j0 + n;
      float pt = -0.5f * hw *
                 (qn[(size_t)h * NRES + gi] + kn[(size_t)h * NRES + gj] - 2.f * c2[r]);
      float bi = lds_bias[(row * 16 + n) * 16 + h];
      float mg = INFV * (mask[gi] * mask[gj] - 1.0f);
      abuf[((size_t)h * NRES + gi) * NRES + gj] = s1 * c1[r] + s2 * bi + pt + mg;
    }
  }
}

// ============================================================ K5b: softmax
__global__ void k_softmax(float* __restrict__ abuf) {
  __shared__ float red[256];
  int t = threadIdx.x;
  float* p = abuf + (size_t)blockIdx.x * NRES;
  float mx = -3.4e38f;
  for (int c = t; c < NRES; c += 256) mx = fmaxf(mx, p[c]);
  red[t] = mx; __syncthreads();
  for (int s = 128; s > 0; s >>= 1) {
    if (t < s) red[t] = fmaxf(red[t], red[t + s]);
    __syncthreads();
  }
  mx = red[0]; __syncthreads();
  float sm = 0.f;
  for (int c = t; c < NRES; c += 256) { float e = expf(p[c] - mx); p[c] = e; sm += e; }
  red[t] = sm; __syncthreads();
  for (int s = 128; s > 0; s >>= 1) {
    if (t < s) red[t] += red[t + s];
    __syncthreads();
  }
  float inv = 1.0f / red[0];
  for (int c = t; c < NRES; c += 256) p[c] *= inv;
}

// ===================================================== K6: o = a@v, o_pt = a@vg
__global__ void k_av(const float* __restrict__ abuf,
                     const float* __restrict__ vh, const float* __restrict__ vg,
                     const float* __restrict__ Rb, const float* __restrict__ tb,
                     float* __restrict__ cat) {
  __shared__ float lpt[16 * 24];
  int i0 = blockIdx.x * 16, h = blockIdx.y;
  int lane = threadIdx.x & 31, m = lane & 15, half = lane >> 4, n = m;
  v8f co = zero8(), cp0 = zero8(), cp1 = zero8();
  const float* arow = abuf + ((size_t)h * NRES + i0 + m) * NRES;
  for (int j0 = 0; j0 < NRES; j0 += 16) {
    for (int k4 = 0; k4 < 16; k4 += 4) {
      int kk = j0 + k4;
      v2f a = mk2(arow[kk + 2 * half], arow[kk + 2 * half + 1]);
      const float* v0 = vh + ((size_t)h * NRES + kk + 2 * half) * CIPA;
      const float* v1 = vh + ((size_t)h * NRES + kk + 2 * half + 1) * CIPA;
      co = wmma4(a, mk2(v0[n], v1[n]), co);
      const float* g0 = vg + ((size_t)h * NRES + kk + 2 * half) * 24;
      const float* g1 = vg + ((size_t)h * NRES + kk + 2 * half + 1) * 24;
      cp0 = wmma4(a, mk2(g0[n], g1[n]), cp0);
      int n2 = n + 16;
      cp1 = wmma4(a, (n2 < 24) ? mk2(g0[n2], g1[n2]) : mk2(0.f, 0.f), cp1);
    }
  }
  #pragma unroll
  for (int r = 0; r < 8; ++r) {
    int row = r + 8 * half, gi = i0 + row;
    cat[(size_t)gi * CATW + h * 16 + n] = co[r];
    lpt[row * 24 + n] = cp0[r];
    if (n + 16 < 24) lpt[row * 24 + n + 16] = cp1[r];
  }
  __syncthreads();
  for (int idx = lane; idx < 16 * PV; idx += 32) {
    int row = idx >> 3, p = idx & 7, gi = i0 + row;
    const float* R = Rb + gi * 9;
    float gx = lpt[row * 24 + p * 3 + 0] - tb[gi * 3 + 0];
    float gy = lpt[row * 24 + p * 3 + 1] - tb[gi * 3 + 1];
    float gz = lpt[row * 24 + p * 3 + 2] - tb[gi * 3 + 2];
    float lx = R[0] * gx + R[3] * gy + R[6] * gz;
    float ly = R[1] * gx + R[4] * gy + R[7] * gz;
    float lz = R[2] * gx + R[5] * gy + R[8] * gz;
    float nm = sqrtf(lx * lx + ly * ly + lz * lz + 1e-8f);
    cat[(size_t)gi * CATW + 192 + h * PV + p] = lx;
    cat[(size_t)gi * CATW + 288 + h * PV + p] = ly;
    cat[(size_t)gi * CATW + 384 + h * PV + p] = lz;
    cat[(size_t)gi * CATW + 480 + h * PV + p] = nm;
  }
}

// ===================================================== K7: o_pair = a @ z[i]
// Double-buffered async-DMA staging of z[i] (the 512MB stream) into LDS.
__global__ void k_opair(const float* __restrict__ abuf, const float* __restrict__ z,
                        float* __restrict__ cat) {
  __shared__ float zbuf[2 * 16 * CZ];   // 2 x 8KB chunks (16 j-rows x 128 ch)
  int i = blockIdx.x;
  int t = threadIdx.x;
  int wv = t >> 5, c0 = wv * 16;
  int lane = t & 31, m = lane & 15, half = lane >> 4, n = m;
  bool valid = (m < NH);
  const float* arow = abuf + ((size_t)(valid ? m : 0) * NRES + i) * NRES;
  const char* zbase = (const char*)(z + (size_t)i * NRES * CZ);
  unsigned lds_base = lds_off_of(&zbuf[0]);

  // issue chunk 0: 512 x 16B packets, 2 per thread
  #pragma unroll
  for (int u = 0; u < 2; ++u) {
    int f = t + 256 * u;
    async_ld_b128(lds_base + (unsigned)f * 16, zbase + f * 16);
  }

  v8f acc = zero8();
  for (int c = 0; c < NRES / 16; ++c) {
    int sel = c & 1;
    if (c + 1 < NRES / 16) {
      const char* gsrc = zbase + (size_t)(c + 1) * 16 * CZ * 4;
      unsigned lbase = lds_base + (unsigned)((c + 1) & 1) * (16 * CZ * 4);
      #pragma unroll
      for (int u = 0; u < 2; ++u) {
        int f = t + 256 * u;
        async_ld_b128(lbase + (unsigned)f * 16, gsrc + f * 16);
      }
      wait_async2();   // chunk c complete, chunk c+1 (2 insts/thread) in flight
    } else {
      wait_async0();
    }
    __syncthreads();   // chunk c visible to all waves

    const float* zs = &zbuf[sel * 16 * CZ];
    int j0 = c * 16;
    #pragma unroll
    for (int kl = 0; kl < 16; kl += 4) {
      v2f a = valid ? mk2(arow[j0 + kl + 2 * half], arow[j0 + kl + 2 * half + 1])
                    : mk2(0.f, 0.f);
      v2f b = mk2(zs[(kl + 2 * half) * CZ + c0 + n],
                  zs[(kl + 2 * half + 1) * CZ + c0 + n]);
      acc = wmma4(a, b, acc);
    }
    __syncthreads();   // all waves done reading chunk c before its buffer reuse
  }
  #pragma unroll
  for (int r = 0; r < 8; ++r) {
    int hrow = r + 8 * half;
    if (hrow < NH)
      cat[(size_t)i * CATW + 576 + hrow * CZ + c0 + n] = acc[r];
  }
}

// ================================================ K8: output projection GEMM
__global__ void k_outproj(const float* __restrict__ cat,
                          const float* __restrict__ wout, const float* __restrict__ bout,
                          const float* __restrict__ s_init, float* __restrict__ sres) {
  int i0 = blockIdx.x * 16, c0 = blockIdx.y * 16;
  int lane = threadIdx.x & 31, m = lane & 15, half = lane >> 4, n = m;
  const float* Arow = cat + (size_t)(i0 + m) * CATW;
  v8f acc = zero8();
  for (int k = 0; k < CATW; k += 4) {
    v2f a = mk2(Arow[k + 2 * half], Arow[k + 2 * half + 1]);
    v2f b = mk2(wout[(size_t)(k + 2 * half) * CS + c0 + n],
                wout[(size_t)(k + 2 * half + 1) * CS + c0 + n]);
    acc = wmma4(a, b, acc);
  }
  #pragma unroll
  for (int r = 0; r < 8; ++r) {
    int row = r + 8 * half, gi = i0 + row;
    sres[(size_t)gi * CS + c0 + n] =
        acc[r] + bout[c0 + n] + s_init[(size_t)gi * CS + c0 + n];
  }
}

// ======================================================= K9: final layernorm
__global__ void k_ln(const float* __restrict__ sres,
                     const float* __restrict__ g, const float* __restrict__ b,
                     float* __restrict__ sout) {
  __shared__ float red[256];
  int t = threadIdx.x;
  const float* p = sres + (size_t)blockIdx.x * CS;
  float s = 0.f;
  for (int c = t; c < CS; c += 256) s += p[c];
  red[t] = s; __syncthreads();
  for (int o = 128; o > 0; o >>= 1) { if (t < o) red[t] += red[t + o]; __syncthreads(); }
  float mean = red[0] / (float)CS; __syncthreads();
  float q = 0.f;
  for (int c = t; c < CS; c += 256) { float d = p[c] - mean; q += d * d; }
  red[t] = q; __syncthreads();
  for (int o = 128; o > 0; o >>= 1) { if (t < o) red[t] += red[t + o]; __syncthreads(); }
  float inv = rsqrtf(red[0] / (float)CS + 1e-5f);
  float* outp = sout + (size_t)blockIdx.x * CS;
  for (int c = t; c < CS; c += 256) outp[c] = (p[c] - mean) * inv * g[c] + b[c];
}

// ===========================================================================
extern "C" void kernel_launch(void* const* d_in, const int* in_sizes, int n_in,
                              void* d_out, int out_size, void* d_ws, size_t ws_size,
                              hipStream_t stream) {
  (void)in_sizes; (void)n_in; (void)out_size; (void)ws_size;
  const int*   ridx  = (const int*)d_in[0];
  const float* coord = (const float*)d_in[1];
  const float* mask  = (const float*)d_in[2];
  const float* rp_w  = (const float*)d_in[3];
  const float* rp_b  = (const float*)d_in[4];
  const float* rd_w  = (const float*)d_in[5];
  const float* rd_b  = (const float*)d_in[6];
  const float* plg   = (const float*)d_in[7];
  const float* plb   = (const float*)d_in[8];
  const float* slg   = (const float*)d_in[9];
  const float* slb   = (const float*)d_in[10];
  const float* wq    = (const float*)d_in[11];
  const float* bq    = (const float*)d_in[12];
  const float* wkv   = (const float*)d_in[13];
  const float* bkv   = (const float*)d_in[14];
  const float* wqp   = (const float*)d_in[15];
  const float* bqp   = (const float*)d_in[16];
  const float* wkp   = (const float*)d_in[17];
  const float* bkp   = (const float*)d_in[18];
  const float* wb    = (const float*)d_in[19];
  const float* bb    = (const float*)d_in[20];
  const float* hwv   = (const float*)d_in[21];
  const float* wout  = (const float*)d_in[22];
  const float* bout  = (const float*)d_in[23];

  float* out_s    = (float*)d_out;                        // 1024*384
  float* out_pair = (float*)d_out + (size_t)NRES * CS;    // 1024*1024*128

  // workspace layout (floats)
  float* w = (float*)d_ws;
  size_t off = 0;
  float* s_init = w + off; off += (size_t)NRES * CS;
  float* Rb     = w + off; off += (size_t)NRES * 9;
  float* tb     = w + off; off += (size_t)NRES * 3;
  float* proj   = w + off; off += (size_t)NRES * PROJW;
  float* qh     = w + off; off += (size_t)NH * NRES * CIPA;
  float* kh     = w + off; off += (size_t)NH * NRES * CIPA;
  float* vh     = w + off; off += (size_t)NH * NRES * CIPA;
  float* qg     = w + off; off += (size_t)NH * NRES * 12;
  float* kg     = w + off; off += (size_t)NH * NRES * 12;
  float* vg     = w + off; off += (size_t)NH * NRES * 24;
  float* qn     = w + off; off += (size_t)NH * NRES;
  float* kn     = w + off; off += (size_t)NH * NRES;
  float* cat    = w + off; off += (size_t)NRES * CATW;
  float* sres   = w + off; off += (size_t)NRES * CS;
  float* abuf   = w + off; off += (size_t)NH * NRES * NRES;   // total ~72 MB

  k_pair<<<(NRES * NRES) / 8, 256, 0, stream>>>(ridx, coord, mask, rp_w, rp_b,
                                                rd_w, rd_b, plg, plb, out_pair);
  k_sin_frames<<<NRES, 192, 0, stream>>>(ridx, coord, s_init, Rb, tb);
  k_proj<<<dim3(NRES / 16, PROJW / 16), 32, 0, stream>>>(
      s_init, wq, bq, wkv, bkv, wqp, bqp, wkp, bkp, proj);
  k_split<<<NRES, 192, 0, stream>>>(proj, Rb, tb, qh, kh, vh, qg, kg, vg, qn, kn);
  k_logits<<<dim3(NRES / 16, NRES / 16), 128, 0, stream>>>(
      qh, kh, qg, kg, qn, kn, out_pair, wb, bb, hwv, mask, abuf);
  k_softmax<<<NH * NRES, 256, 0, stream>>>(abuf);
  k_av<<<dim3(NRES / 16, NH), 32, 0, stream>>>(abuf, vh, vg, Rb, tb, cat);
  k_opair<<<NRES, 256, 0, stream>>>(abuf, out_pair, cat);
  k_outproj<<<dim3(NRES / 16, CS / 16), 32, 0, stream>>>(cat, wout, bout, s_init, sres);
  k_ln<<<NRES, 256, 0, stream>>>(sres, slg, slb, out_s);
}